// GSAGE_57397942943812
// MI455X (gfx1250) — compile-verified
//
#include <hip/hip_runtime.h>

// GraphSAGE layer on MI455X (gfx1250, wave32).
//   agg[i]  = (1/max(deg_i,1)) * sum_{e: row[e]==i} val[e] * x[col[e]]
//   out     = relu(agg @ W_l^T + b_l + x @ W_r^T)
// Phase 1: edge scatter with f32 atomics (x + agg are L2-resident: 51 MB each vs 192 MB L2).
// Phase 2: fused dual-GEMM epilogue using V_WMMA_F32_16X16X4_F32 (full fp32 precision).

typedef __attribute__((ext_vector_type(2))) float v2f;
typedef __attribute__((ext_vector_type(8))) float v8f;

#define N_NODES 100000
#define N_EDGES 1600000
#define F 128            // F_IN == F_OUT == 128

// ---------------------------------------------------------------------------
// K0: zero workspace (agg + deg). Harness poisons ws once; we must re-zero
// every call since kernel_launch must be deterministic.
// ---------------------------------------------------------------------------
__global__ void GSAGE_zero_kernel(float* __restrict__ p, long n) {
    long i = (long)blockIdx.x * blockDim.x + threadIdx.x;
    if (i < n) p[i] = 0.0f;
}

// ---------------------------------------------------------------------------
// K1: one wave (32 lanes) per edge.
//   lane l loads x[col][4l..4l+3] (float4, coalesced, L2-resident)
//   and atomically accumulates val*x into agg[row][4l..4l+3].
//   lane 0 bumps the degree counter.
// ---------------------------------------------------------------------------
__global__ void __launch_bounds__(256)
GSAGE_scatter_kernel(const float* __restrict__ x, const float* __restrict__ val,
                     const int* __restrict__ row, const int* __restrict__ col,
                     float* __restrict__ agg, float* __restrict__ deg) {
    int e    = (blockIdx.x * blockDim.x + threadIdx.x) >> 5;
    int lane = threadIdx.x & 31;
    if (e >= N_EDGES) return;
    int r = row[e];
    int c = col[e];
    float v = val[e];
    float4 m = *(const float4*)(x + (size_t)c * F + lane * 4);
    float* dst = agg + (size_t)r * F + lane * 4;
    atomicAdd(dst + 0, v * m.x);
    atomicAdd(dst + 1, v * m.y);
    atomicAdd(dst + 2, v * m.z);
    atomicAdd(dst + 3, v * m.w);
    if (lane == 0) atomicAdd(deg + r, 1.0f);
}

// ---------------------------------------------------------------------------
// K2: deg -> 1/max(deg,1)  (in place)
// ---------------------------------------------------------------------------
__global__ void GSAGE_rdeg_kernel(float* __restrict__ deg, int n) {
    int i = blockIdx.x * blockDim.x + threadIdx.x;
    if (i < n) deg[i] = 1.0f / fmaxf(deg[i], 1.0f);
}

// ---------------------------------------------------------------------------
// K3: fused out = relu((agg*rdeg) @ W_l^T + b_l + x @ W_r^T) via f32 WMMA.
//
// Fragment layouts per CDNA5 ISA 7.12.2 (wave32):
//   A 16x4 f32 (2 VGPRs): lane L (0-15): M=L, holds K={0,1}; lanes 16-31: K={2,3}.
//   B 4x16  f32 (2 VGPRs): lane L (0-15): N=L, rows K={0,1}; lanes 16-31: K={2,3}.
//   C/D 16x16 f32 (8 VGPRs): VGPR r = row M=r (lanes 0-15) / M=r+8 (lanes 16-31),
//                            N = lane%16.
//
// Block = 256 threads = 8 waves; blockIdx.x selects 16 node rows (100000/16 =
// 6250 blocks exactly -> EXEC all-ones, as WMMA requires); wave id selects the
// 16-column output tile. Both GEMMs chain into one f32 accumulator.
// ---------------------------------------------------------------------------
__global__ void __launch_bounds__(256)
GSAGE_gemm_kernel(const float* __restrict__ agg, const float* __restrict__ rdeg,
                  const float* __restrict__ x,   const float* __restrict__ Wl,
                  const float* __restrict__ bl,  const float* __restrict__ Wr,
                  float* __restrict__ out) {
    const int nodeBase = blockIdx.x * 16;
    const int wave  = threadIdx.x >> 5;        // 0..7: output column tile
    const int lane  = threadIdx.x & 31;
    const int half  = lane >> 4;               // 0: K={k,k+1}  1: K={k+2,k+3}
    const int l16   = lane & 15;               // A row / B,D col within tile
    const int jBase = wave * 16;

    const float* aggRow = agg + (size_t)(nodeBase + l16) * F;
    const float* xRow   = x   + (size_t)(nodeBase + l16) * F;
    const float  rd     = rdeg[nodeBase + l16];
    // B[k][j] = W[j][k]; W stored row-major [F_OUT, F_IN] -> column j is a
    // contiguous 128-float row of W, so float2 loads are unit-stride.
    const float* wlCol  = Wl + (size_t)(jBase + l16) * F;
    const float* wrCol  = Wr + (size_t)(jBase + l16) * F;

    v8f c = {};
    #pragma unroll 4
    for (int kb = 0; kb < F; kb += 4) {
        const int k = kb + 2 * half;
        float2 t;
        v2f a_agg, a_x, b_l, b_r;
        t = *(const float2*)(aggRow + k); a_agg[0] = t.x * rd; a_agg[1] = t.y * rd;
        t = *(const float2*)(xRow + k);   a_x[0]   = t.x;      a_x[1]   = t.y;
        t = *(const float2*)(wlCol + k);  b_l[0]   = t.x;      b_l[1]   = t.y;
        t = *(const float2*)(wrCol + k);  b_r[0]   = t.x;      b_r[1]   = t.y;
        c = __builtin_amdgcn_wmma_f32_16x16x4_f32(false, a_agg, false, b_l,
                                                  (short)0, c, false, false);
        c = __builtin_amdgcn_wmma_f32_16x16x4_f32(false, a_x,   false, b_r,
                                                  (short)0, c, false, false);
    }

    const float bias = bl[jBase + l16];
    #pragma unroll
    for (int r = 0; r < 8; ++r) {
        int mr = half ? (r + 8) : r;
        float v = c[r] + bias;
        out[(size_t)(nodeBase + mr) * F + jBase + l16] = fmaxf(v, 0.0f);
    }
}

// ---------------------------------------------------------------------------
extern "C" void kernel_launch(void* const* d_in, const int* in_sizes, int n_in,
                              void* d_out, int out_size, void* d_ws, size_t ws_size,
                              hipStream_t stream) {
    const float* x   = (const float*)d_in[0];
    const float* val = (const float*)d_in[1];
    const float* Wl  = (const float*)d_in[2];
    const float* bl  = (const float*)d_in[3];
    const float* Wr  = (const float*)d_in[4];
    const int*   row = (const int*)d_in[5];
    const int*   col = (const int*)d_in[6];
    float*       out = (float*)d_out;

    float* agg = (float*)d_ws;                       // N*F floats (51.2 MB)
    float* deg = agg + (size_t)N_NODES * F;          // N floats

    // K0: zero agg + deg
    long zn = (long)N_NODES * F + N_NODES;
    GSAGE_zero_kernel<<<(int)((zn + 255) / 256), 256, 0, stream>>>(agg, zn);

    // K1: edge scatter (one wave per edge -> E/8 blocks of 256)
    GSAGE_scatter_kernel<<<N_EDGES / 8, 256, 0, stream>>>(x, val, row, col, agg, deg);

    // K2: reciprocal degree
    GSAGE_rdeg_kernel<<<(N_NODES + 255) / 256, 256, 0, stream>>>(deg, N_NODES);

    // K3: fused WMMA dual-GEMM + bias + relu
    GSAGE_gemm_kernel<<<N_NODES / 16, 256, 0, stream>>>(agg, deg, x, Wl, bl, Wr, out);
}